// Equiformer_72808285602181
// MI455X (gfx1250) — compile-verified
//
#include <hip/hip_runtime.h>
#include <math.h>

// ---------------- model constants ----------------
#define NNODES 1024
#define NEDGES 8192
#define NGRAPH 16
#define NBASIS 32
#define CUTOFF_R 5.0f
#define NHEADS 4
#define NLAYERS 6
#define MAXAT 10
#define WNUM 960

// 15 tensor-product paths (l1,l2,l3) in reference order
__constant__ int PATH_L1[15]   = {0,0,0,1,1,1,1,1,1,2,2,2,2,2,2};
__constant__ int PATH_L2[15]   = {0,1,2,0,1,1,1,2,2,0,1,1,2,2,2};
__constant__ int PATH_L3[15]   = {0,1,2,1,0,1,2,1,2,2,1,2,0,1,2};
__constant__ int PATH_WOFF[15] = {0,128,256,384,448,512,576,640,704,768,800,832,864,896,928};
__constant__ int PATH_COFF[15] = {0,0,0,128,128,192,128,256,192,256,320,288,192,352,320};
__constant__ int PATH_CGOFF[15]= {0,1,10,35,44,53,80,125,170,245,270,315,390,415,490};
__constant__ int MUL_C[3]  = {128,64,32};   // NODE_MUL
__constant__ int DTPO_C[3] = {224,384,352}; // DTPO_MUL

// ---------------- WMMA / async-LDS setup ----------------
typedef __attribute__((ext_vector_type(8))) float v8f;
typedef __attribute__((ext_vector_type(2))) float v2f;
typedef int v4i __attribute__((vector_size(16)));
#define AS1 __attribute__((address_space(1)))
#define AS3 __attribute__((address_space(3)))

#if defined(__AMDGCN__) && defined(__has_builtin)
#  if __has_builtin(__builtin_amdgcn_global_load_async_to_lds_b128) && \
      __has_builtin(__builtin_amdgcn_s_wait_asynccnt)
#    define USE_ASYNC_LDS 1
#  endif
#endif

#if defined(USE_ASYNC_LDS)
// 16B global -> LDS async copy (tracked by ASYNCcnt)
__device__ __forceinline__ void async_copy16(const float* g, float* l) {
  __builtin_amdgcn_global_load_async_to_lds_b128(
      (AS1 v4i*)(float*)g, (AS3 v4i*)l, 0, 0);
}
#endif

// Generic row-major GEMM: C[R,N] (+)= (A[R,K] @ W[K,N]) * scale
// 128 threads = 4 waves per workgroup; 32x32 macro-tile; K staged through LDS
// in 32-wide panels; each wave owns one 16x16 sub-tile and issues 8 WMMAs/panel.
// Requirements (all satisfied here): R%32==0, N%32==0, K%32==0.
#define TM 32
#define TN 32
#define KC 32

__global__ __launch_bounds__(128)
void gemm_wmma_kernel(const float* __restrict__ A, const float* __restrict__ B,
                      float* __restrict__ C, int R, int K, int N, float scale, int addOut)
{
  __shared__ __align__(16) float As[TM * KC];  // [m][k]
  __shared__ __align__(16) float Bs[KC * TN];  // [k][n]
  const int tid  = threadIdx.x;
  const int wave = tid >> 5;
  const int lane = tid & 31;
  const int half = lane >> 4;
  const int l16  = lane & 15;
  const int tile_m = blockIdx.x * TM;
  const int tile_n = blockIdx.y * TN;
  const int wm = (wave >> 1) * 16;
  const int wn = (wave & 1) * 16;
  // cooperative staging: each thread moves 8 floats of A and 8 floats of B per panel
  const int srow = tid >> 2;           // 0..31
  const int scol = (tid & 3) * 8;      // 0,8,16,24

  v8f acc = {};
  for (int k0 = 0; k0 < K; k0 += KC) {
    const float* gA = A + (size_t)(tile_m + srow) * K + k0 + scol;
    const float* gB = B + (size_t)(k0 + srow) * N + tile_n + scol;
    if (k0 + KC < K) {  // prefetch next panels into L2 (global_prefetch_b8)
      __builtin_prefetch(gA + KC, 0, 1);
      __builtin_prefetch(gB + (size_t)KC * N, 0, 1);
    }
#if defined(USE_ASYNC_LDS)
    async_copy16(gA,     &As[srow * KC + scol]);
    async_copy16(gA + 4, &As[srow * KC + scol + 4]);
    async_copy16(gB,     &Bs[srow * TN + scol]);
    async_copy16(gB + 4, &Bs[srow * TN + scol + 4]);
    __builtin_amdgcn_s_wait_asynccnt(0);
#else
    const float4 va0 = *(const float4*)gA;
    const float4 va1 = *(const float4*)(gA + 4);
    const float4 vb0 = *(const float4*)gB;
    const float4 vb1 = *(const float4*)(gB + 4);
    *(float4*)&As[srow * KC + scol]     = va0;
    *(float4*)&As[srow * KC + scol + 4] = va1;
    *(float4*)&Bs[srow * TN + scol]     = vb0;
    *(float4*)&Bs[srow * TN + scol + 4] = vb1;
#endif
    __syncthreads();
#pragma unroll
    for (int kk = 0; kk < KC; kk += 4) {
      // A 16x4 f32: lanes0-15 -> K pair (kk,kk+1), lanes16-31 -> (kk+2,kk+3), M=lane&15
      v2f a, b;
      const int ka = kk + half * 2;
      a.x = As[(wm + l16) * KC + ka];
      a.y = As[(wm + l16) * KC + ka + 1];
      b.x = Bs[ka * TN + wn + l16];
      b.y = Bs[(ka + 1) * TN + wn + l16];
      acc = __builtin_amdgcn_wmma_f32_16x16x4_f32(false, a, false, b, (short)0, acc,
                                                  false, false);
    }
    __syncthreads();
  }
  // C/D layout (ISA): VGPR r -> (M = half*8 + r, N = lane&15)
#pragma unroll
  for (int r = 0; r < 8; ++r) {
    const int row = tile_m + wm + half * 8 + r;
    const size_t idx = (size_t)row * N + tile_n + wn + l16;
    const float v = acc[r] * scale;
    if (addOut) C[idx] += v; else C[idx] = v;
  }
}

// ---------------- Clebsch-Gordan tables (computed on device, double precision) ----
__device__ double factd(int n) { double r = 1.0; for (int i = 2; i <= n; ++i) r *= i; return r; }

__device__ double su2_cg(int j1, int m1, int j2, int m2, int j3, int m3) {
  if (m3 != m1 + m2) return 0.0;
  double pref = (2.0 * j3 + 1.0) * factd(j1 + j2 - j3) * factd(j1 - j2 + j3) *
                factd(-j1 + j2 + j3) / factd(j1 + j2 + j3 + 1);
  pref = sqrt(pref * factd(j1 + m1) * factd(j1 - m1) * factd(j2 + m2) *
              factd(j2 - m2) * factd(j3 + m3) * factd(j3 - m3));
  double s = 0.0;
  for (int k = 0; k <= j1 + j2 - j3; ++k) {
    int dn[6] = {k, j1 + j2 - j3 - k, j1 - m1 - k, j2 + m2 - k,
                 j3 - j2 + m1 + k, j3 - j1 - m2 + k};
    bool ok = true; double den = 1.0;
    for (int t = 0; t < 6; ++t) { if (dn[t] < 0) { ok = false; break; } den *= factd(dn[t]); }
    if (!ok) continue;
    s += ((k & 1) ? -1.0 : 1.0) / den;
  }
  return pref * s;
}

__device__ void build_q(int l, double qr[5][5], double qi[5][5]) {
  for (int a = 0; a < 5; ++a) for (int b = 0; b < 5; ++b) { qr[a][b] = 0; qi[a][b] = 0; }
  const double is2 = 1.0 / sqrt(2.0);
  for (int m = -l; m < 0; ++m) {
    qr[l + m][l - m] = is2;     // q[l+m, l+|m|] = 1/sqrt2
    qi[l + m][l + m] = -is2;    // q[l+m, l-|m|] = -i/sqrt2
  }
  qr[l][l] = 1.0;
  for (int m = 1; m <= l; ++m) {
    const double sgn = (m & 1) ? -1.0 : 1.0;
    qr[l + m][l + m] = sgn * is2;
    qi[l + m][l - m] = sgn * is2;   // i*(-1)^m/sqrt2
  }
  double fr, fi;  // (-i)^l
  if (l == 0) { fr = 1; fi = 0; } else if (l == 1) { fr = 0; fi = -1; } else { fr = -1; fi = 0; }
  for (int a = 0; a < 2 * l + 1; ++a)
    for (int b = 0; b < 2 * l + 1; ++b) {
      const double r = qr[a][b], i = qi[a][b];
      qr[a][b] = r * fr - i * fi;
      qi[a][b] = r * fi + i * fr;
    }
}

__global__ void cg_kernel(float* __restrict__ cg) {
  const int p = threadIdx.x;
  if (p >= 15) return;
  const int l1 = PATH_L1[p], l2 = PATH_L2[p], l3 = PATH_L3[p];
  double q1r[5][5], q1i[5][5], q2r[5][5], q2i[5][5], q3r[5][5], q3i[5][5];
  build_q(l1, q1r, q1i); build_q(l2, q2r, q2i); build_q(l3, q3r, q3i);
  const int n1 = 2 * l1 + 1, n2 = 2 * l2 + 1, n3 = 2 * l3 + 1;
  float* o = cg + PATH_CGOFF[p];
  for (int i = 0; i < n1; ++i)
    for (int j = 0; j < n2; ++j)
      for (int k = 0; k < n3; ++k) {
        double acc = 0.0;
        for (int a = 0; a < n1; ++a)
          for (int b = 0; b < n2; ++b) {
            const int m1 = a - l1, m2 = b - l2, m3 = m1 + m2;
            if (m3 < -l3 || m3 > l3) continue;
            const double cgv = su2_cg(l1, m1, l2, m2, l3, m3);
            if (cgv == 0.0) continue;
            const int c = l3 + m3;
            const double ar = q1r[a][i], ai = q1i[a][i];
            const double br = q2r[b][j], bi = q2i[b][j];
            const double cr = q3r[c][k], ci = -q3i[c][k]; // conj
            const double tr = ar * br - ai * bi, ti = ar * bi + ai * br;
            acc += (tr * cr - ti * ci) * cgv;
          }
        o[(i * n2 + j) * n3 + k] = (float)acc;
      }
}

// ---------------- geometry: vec, bessel RBF, spherical harmonics ----------------
__global__ void geom_kernel(const float* __restrict__ pos, const int* __restrict__ srcI,
                            const int* __restrict__ dstI, float* __restrict__ rbf,
                            float* __restrict__ sh1, float* __restrict__ sh2, int E)
{
  const int e = blockIdx.x * blockDim.x + threadIdx.x;
  if (e >= E) return;
  const int s = srcI[e], d = dstI[e];
  // p3 = pos[:, [1,2,0]]
  const float vx = pos[d * 3 + 1] - pos[s * 3 + 1];
  const float vy = pos[d * 3 + 2] - pos[s * 3 + 2];
  const float vz = pos[d * 3 + 0] - pos[s * 3 + 0];
  const float dist = sqrtf(vx * vx + vy * vy + vz * vz);
  float u = fminf(fmaxf(dist / CUTOFF_R, 0.f), 1.f);
  const float u5 = u * u * u * u * u;
  const float fc = 1.f - 21.f * u5 + 35.f * u5 * u - 15.f * u5 * u * u;
  const float pref = sqrtf(2.f / CUTOFF_R) / (dist + 1e-9f) * fc;
  const float pi = 3.14159265358979323846f;
  for (int n = 1; n <= NBASIS; ++n)
    rbf[(size_t)e * NBASIS + n - 1] = pref * sinf((float)n * pi * dist / CUTOFF_R);
  const float inv = 1.f / (dist + 1e-9f);
  const float nx = vx * inv, ny = vy * inv, nz = vz * inv;
  const float s3 = sqrtf(3.f), s15 = sqrtf(15.f), s5 = sqrtf(5.f);
  sh1[e * 3 + 0] = s3 * ny;
  sh1[e * 3 + 1] = s3 * nz;
  sh1[e * 3 + 2] = s3 * nx;
  sh2[e * 5 + 0] = s15 * nx * ny;
  sh2[e * 5 + 1] = s15 * ny * nz;
  sh2[e * 5 + 2] = 0.5f * s5 * (3.f * nz * nz - 1.f);
  sh2[e * 5 + 3] = s15 * nx * nz;
  sh2[e * 5 + 4] = 0.5f * s15 * (nx * nx - ny * ny);
}

// ---------------- small elementwise / init kernels ----------------
__global__ void fill_f_kernel(float* p, float v, size_t n) {
  size_t t = (size_t)blockIdx.x * blockDim.x + threadIdx.x;
  if (t < n) p[t] = v;
}
__global__ void fill_u_kernel(unsigned int* p, unsigned int v, size_t n) {
  size_t t = (size_t)blockIdx.x * blockDim.x + threadIdx.x;
  if (t < n) p[t] = v;
}
__global__ void one0_kernel(const float* eW, const float* eb, float* one0) {
  const int u = threadIdx.x;
  if (u < 128) one0[u] = eW[u] + eb[u];
}
__global__ void node_embed_kernel(const int* __restrict__ at_no, const float* __restrict__ atom_W,
                                  const float* __restrict__ atom_b, float* __restrict__ node0, int N)
{
  const int t = blockIdx.x * blockDim.x + threadIdx.x;
  if (t >= N * 128) return;
  const int n = t >> 7, u = t & 127;
  node0[t] = atom_W[at_no[n] * 128 + u] * (1.f / sqrtf((float)MAXAT)) + atom_b[u];
}
__global__ void sigmoid_kernel(float* x, size_t n) {
  size_t t = (size_t)blockIdx.x * blockDim.x + threadIdx.x;
  if (t < n) x[t] = 1.f / (1.f + expf(-x[t]));
}
// x layout [nm][R][mul]; per (row,u): scale all nm components by sigmoid(norm)
__global__ void gate_vec_kernel(float* __restrict__ x, int R, int mul, int nm) {
  const int t = blockIdx.x * blockDim.x + threadIdx.x;
  if (t >= R * mul) return;
  const int e = t / mul, u = t - e * mul;
  float s2 = 1e-12f;
  for (int m = 0; m < nm; ++m) {
    const float v = x[((size_t)m * R + e) * mul + u];
    s2 += v * v;
  }
  const float sg = 1.f / (1.f + expf(-sqrtf(s2)));
  for (int m = 0; m < nm; ++m) {
    const size_t idx = ((size_t)m * R + e) * mul + u;
    x[idx] *= sg;
  }
}

// ---------------- equivariant layernorm ----------------
__global__ void o3_ln_kernel(const float* __restrict__ x, float* __restrict__ out,
                             const float* __restrict__ g, const float* __restrict__ b,
                             int N, int mul, int nm, int isL0)
{
  __shared__ float red[128];
  const int n = blockIdx.x, t = threadIdx.x;
  const int cnt = mul * nm;
  float mean = 0.f;
  if (isL0) {
    float s = 0.f;
    for (int i = t; i < mul; i += 128) s += x[(size_t)n * mul + i];
    red[t] = s; __syncthreads();
    for (int o = 64; o > 0; o >>= 1) { if (t < o) red[t] += red[t + o]; __syncthreads(); }
    mean = red[0] / (float)mul;
    __syncthreads();
  }
  float s2 = 0.f;
  for (int i = t; i < cnt; i += 128) {
    const int m = i / mul, u = i - m * mul;
    const float v = x[((size_t)m * N + n) * mul + u] - (isL0 ? mean : 0.f);
    s2 += v * v;
  }
  red[t] = s2; __syncthreads();
  for (int o = 64; o > 0; o >>= 1) { if (t < o) red[t] += red[t + o]; __syncthreads(); }
  const float inv = rsqrtf(red[0] / (float)cnt + 1e-5f);
  for (int i = t; i < cnt; i += 128) {
    const int m = i / mul, u = i - m * mul;
    const size_t idx = ((size_t)m * N + n) * mul + u;
    float o2 = (x[idx] - (isL0 ? mean : 0.f)) * inv * g[u];
    if (isL0) o2 += b[u];
    out[idx] = o2;
  }
}

// ---------------- depthwise tensor product (15 paths via blockIdx.y) ------------
__global__ void dtp_kernel(const float* __restrict__ x0, const float* __restrict__ x1,
                           const float* __restrict__ x2,
                           const float* __restrict__ sh1, const float* __restrict__ sh2,
                           const float* __restrict__ w, int w_per_edge, int bcast0,
                           const float* __restrict__ cg,
                           float* __restrict__ m0, float* __restrict__ m1, float* __restrict__ m2,
                           int E)
{
  const int p = blockIdx.y;
  const int l1 = PATH_L1[p], l2 = PATH_L2[p], l3 = PATH_L3[p];
  const int mul = MUL_C[l1];
  const int t = blockIdx.x * blockDim.x + threadIdx.x;
  if (t >= E * mul) return;
  const int e = t / mul, u = t - e * mul;
  const int n1 = 2 * l1 + 1, n2 = 2 * l2 + 1, n3 = 2 * l3 + 1;
  const float* xp = (l1 == 0) ? x0 : (l1 == 1) ? x1 : x2;
  float xv[5];
  for (int i = 0; i < n1; ++i) {
    if (!xp) { xv[i] = 0.f; continue; }
    const size_t idx = (bcast0 && l1 == 0) ? (size_t)u : ((size_t)i * E + e) * mul + u;
    xv[i] = xp[idx];
  }
  float shv[5];
  if (l2 == 0) shv[0] = 1.f;
  else if (l2 == 1) { for (int j = 0; j < 3; ++j) shv[j] = sh1[e * 3 + j]; }
  else { for (int j = 0; j < 5; ++j) shv[j] = sh2[e * 5 + j]; }
  const float wv = w[(w_per_edge ? (size_t)e * WNUM : 0) + PATH_WOFF[p] + u];
  const float* cgp = cg + PATH_CGOFF[p];
  float* mo = (l3 == 0) ? m0 : (l3 == 1) ? m1 : m2;
  const int om = DTPO_C[l3], co = PATH_COFF[p];
  for (int k = 0; k < n3; ++k) {
    float acc = 0.f;
    for (int i = 0; i < n1; ++i) {
      const float xi = xv[i];
      for (int j = 0; j < n2; ++j)
        acc += xi * shv[j] * cgp[(i * n2 + j) * n3 + k];
    }
    mo[((size_t)k * E + e) * om + co + u] = acc * wv;
  }
}

// ---------------- gather / scatter ----------------
__global__ void msg_gather_kernel(const float* __restrict__ s, const float* __restrict__ d,
                                  const int* __restrict__ srcI, const int* __restrict__ dstI,
                                  float* __restrict__ msg, int E, int Nn, int mul, int nm)
{
  const int t = blockIdx.x * blockDim.x + threadIdx.x;
  if (t >= nm * E * mul) return;
  const int u = t % mul;
  const int rest = t / mul;
  const int e = rest % E;
  const int m = rest / E;
  msg[t] = s[((size_t)m * Nn + srcI[e]) * mul + u] + d[((size_t)m * Nn + dstI[e]) * mul + u];
}
__global__ void scatter_add_kernel(const float* __restrict__ in, const int* __restrict__ dstI,
                                   float* __restrict__ out, int E, int Nn, int mul, int nm)
{
  const int t = blockIdx.x * blockDim.x + threadIdx.x;
  if (t >= nm * E * mul) return;
  const int u = t % mul;
  const int rest = t / mul;
  const int e = rest % E;
  const int m = rest / E;
  atomicAdd(&out[((size_t)m * Nn + dstI[e]) * mul + u], in[t]);
}
__global__ void att_scatter_kernel(const float* __restrict__ v, const float* __restrict__ aH,
                                   const int* __restrict__ dstI, float* __restrict__ out,
                                   int E, int Nn, int mul, int nm, int hshift)
{
  const int t = blockIdx.x * blockDim.x + threadIdx.x;
  if (t >= nm * E * mul) return;
  const int u = t % mul;
  const int rest = t / mul;
  const int e = rest % E;
  const int m = rest / E;
  const float a = aH[e * NHEADS + (u >> hshift)];
  atomicAdd(&out[((size_t)m * Nn + dstI[e]) * mul + u], v[t] * a);
}

// ---------------- attention softmax pieces ----------------
__device__ inline unsigned int fenc(float f) {
  unsigned int u = __float_as_uint(f);
  return (u & 0x80000000u) ? ~u : (u | 0x80000000u);
}
__device__ inline float fdec(unsigned int e) {
  unsigned int u = (e & 0x80000000u) ? (e ^ 0x80000000u) : ~e;
  return __uint_as_float(u);
}
__global__ void alpha_prep_kernel(const float* __restrict__ aE, const float* __restrict__ adot,
                                  float* __restrict__ aH, int E)
{
  const int t = blockIdx.x * blockDim.x + threadIdx.x;
  if (t >= E * NHEADS) return;
  const int e = t >> 2, h = t & 3;
  float acc = 0.f;
  for (int j = 0; j < 32; ++j) {
    float x = aE[(size_t)e * 128 + h * 32 + j];
    x = (x < 0.f) ? 0.2f * x : x;
    acc += x * adot[h * 32 + j];
  }
  aH[t] = acc;
}
__global__ void amax_kernel(const float* __restrict__ aH, const int* __restrict__ dstI,
                            unsigned int* __restrict__ amax, int E)
{
  const int t = blockIdx.x * blockDim.x + threadIdx.x;
  if (t >= E * NHEADS) return;
  const int e = t >> 2, h = t & 3;
  atomicMax(&amax[dstI[e] * NHEADS + h], fenc(aH[t]));
}
__global__ void ea_kernel(const float* __restrict__ aH, const int* __restrict__ dstI,
                          const unsigned int* __restrict__ amax, float* __restrict__ ea,
                          float* __restrict__ den, int E)
{
  const int t = blockIdx.x * blockDim.x + threadIdx.x;
  if (t >= E * NHEADS) return;
  const int e = t >> 2, h = t & 3;
  float am = fdec(amax[dstI[e] * NHEADS + h]);
  const unsigned int ub = __float_as_uint(am);
  if ((ub & 0x7F800000u) == 0x7F800000u) am = 0.f;  // where(isfinite, amax, 0)
  const float v = expf(aH[t] - am);
  ea[t] = v;
  atomicAdd(&den[dstI[e] * NHEADS + h], v);
}
__global__ void anorm_kernel(const float* __restrict__ ea, const int* __restrict__ dstI,
                             const float* __restrict__ den, float* __restrict__ aH, int E)
{
  const int t = blockIdx.x * blockDim.x + threadIdx.x;
  if (t >= E * NHEADS) return;
  const int e = t >> 2, h = t & 3;
  aH[t] = ea[t] / (den[dstI[e] * NHEADS + h] + 1e-16f);
}

// ---------------- head ----------------
__global__ void head_kernel(const float* __restrict__ y0, const float* __restrict__ lin2,
                            const float* __restrict__ shifts, const int* __restrict__ at_no,
                            const int* __restrict__ batch, float* __restrict__ out, int N)
{
  __shared__ float red[128];
  const int n = blockIdx.x, t = threadIdx.x;
  red[t] = y0[(size_t)n * 128 + t] * lin2[t];
  __syncthreads();
  for (int o = 64; o > 0; o >>= 1) { if (t < o) red[t] += red[t + o]; __syncthreads(); }
  if (t == 0) {
    float v = red[0] * (1.f / sqrtf(128.f)) * (1.f / 18.03065905448718f);
    v += shifts[at_no[n]];
    atomicAdd(&out[batch[n]], v);
  }
}

// ================= host orchestration =================
extern "C" void kernel_launch(void* const* d_in, const int* in_sizes, int n_in,
                              void* d_out, int out_size, void* d_ws, size_t ws_size,
                              hipStream_t stream)
{
  (void)in_sizes; (void)n_in; (void)ws_size;
  const int N = NNODES, E = NEDGES;
  const int* at_no = (const int*)d_in[0];
  const float* pos = (const float*)d_in[1];
  const int* srcI  = (const int*)d_in[2];
  const int* dstI  = srcI + E;
  const int* batch = (const int*)d_in[3];
  auto F = [&](int i) { return (const float*)d_in[i]; };
  // params flattened alphabetically (jax tree order): emb, head, layers
  const float* emb_atom_W = F(4);  const float* emb_atom_b = F(5);
  const float* emb_exp_W  = F(6);  const float* emb_exp_b  = F(7);
  const float* emb_proj[3] = {F(8), F(9), F(10)};
  const float* emb_rbfW = F(11);
  const float* h_lin1_0 = F(12);
  const float* h_lin2   = F(15);
  const float* h_ln_b   = F(16);
  const float* h_ln_g[3] = {F(17), F(18), F(19)};
  const float* h_shifts = F(20);
  auto LPb = [&](int L) { return 21 + L * 33; };

  // workspace carve-up (floats)
  float* base = (float*)d_ws;
  size_t off = 0;
  auto alloc = [&](size_t n) { float* p = base + off; off += (n + 63) & ~(size_t)63; return p; };
  const int NM[3] = {128, 64, 32}, NP[3] = {1, 3, 5}, DT[3] = {224, 384, 352}, HM[3] = {384, 192, 96};
  float* cg = alloc(615);
  float *node[3], *nn[3], *sb[3], *db[3], *att[3], *hid[3], *msg[3], *mb[3], *vout[3];
  for (int l = 0; l < 3; ++l) node[l] = alloc((size_t)NP[l] * N * NM[l]);
  for (int l = 0; l < 3; ++l) nn[l]   = alloc((size_t)NP[l] * N * NM[l]);
  for (int l = 0; l < 3; ++l) sb[l]   = alloc((size_t)NP[l] * N * NM[l]);
  for (int l = 0; l < 3; ++l) db[l]   = alloc((size_t)NP[l] * N * NM[l]);
  for (int l = 0; l < 3; ++l) att[l]  = alloc((size_t)NP[l] * N * NM[l]);
  for (int l = 0; l < 3; ++l) hid[l]  = alloc((size_t)NP[l] * N * HM[l]);
  for (int l = 0; l < 3; ++l) msg[l]  = alloc((size_t)NP[l] * E * NM[l]);  // also "v" after gating
  for (int l = 0; l < 3; ++l) mb[l]   = alloc((size_t)NP[l] * E * DT[l]);  // dtp outputs (reused)
  for (int l = 0; l < 3; ++l) vout[l] = alloc((size_t)NP[l] * E * NM[l]);
  float* rbfB  = alloc((size_t)E * NBASIS);
  float* sh1   = alloc((size_t)E * 3);
  float* sh2   = alloc((size_t)E * 5);
  float* w1    = alloc((size_t)E * WNUM);
  float* one0  = alloc(128);
  float* alphaE = alloc((size_t)E * 128);
  float* aH    = alloc((size_t)E * NHEADS);
  float* eaB   = alloc((size_t)E * NHEADS);
  unsigned int* amax = (unsigned int*)alloc((size_t)N * NHEADS);
  float* den   = alloc((size_t)N * NHEADS);

  auto grid1 = [](size_t n, int b) { return dim3((unsigned)((n + b - 1) / b)); };
  auto gemm = [&](const float* A, const float* Wt, float* C, int R, int K, int Nc, int add) {
    const float sc = 1.0f / sqrtf((float)K);
    gemm_wmma_kernel<<<dim3(R / TM, Nc / TN), 128, 0, stream>>>(A, Wt, C, R, K, Nc, sc, add);
  };
  auto fillf = [&](float* p, float v, size_t n) {
    fill_f_kernel<<<grid1(n, 256), 256, 0, stream>>>(p, v, n);
  };
  auto dtp = [&](const float* x0, const float* x1, const float* x2, const float* w,
                 int wpe, int bc, float* o0, float* o1, float* o2) {
    dtp_kernel<<<dim3((E * 128 + 255) / 256, 15), 256, 0, stream>>>(
        x0, x1, x2, sh1, sh2, w, wpe, bc, cg, o0, o1, o2, E);
  };

  // ---- setup ----
  cg_kernel<<<1, 32, 0, stream>>>(cg);
  geom_kernel<<<grid1(E, 256), 256, 0, stream>>>(pos, srcI, dstI, rbfB, sh1, sh2, E);

  // ---- embedding ----
  node_embed_kernel<<<grid1((size_t)N * 128, 256), 256, 0, stream>>>(at_no, emb_atom_W, emb_atom_b, node[0], N);
  fillf(node[1], 0.f, (size_t)3 * N * 64);
  fillf(node[2], 0.f, (size_t)5 * N * 32);
  one0_kernel<<<1, 128, 0, stream>>>(emb_exp_W, emb_exp_b, one0);
  gemm(rbfB, emb_rbfW, w1, E, NBASIS, WNUM, 0);
  dtp(one0, nullptr, nullptr, w1, /*per-edge*/1, /*bcast0*/1, mb[0], mb[1], mb[2]);
  for (int l = 0; l < 3; ++l) gemm(mb[l], emb_proj[l], msg[l], NP[l] * E, DT[l], NM[l], 0);
  for (int l = 0; l < 3; ++l)
    scatter_add_kernel<<<grid1((size_t)NP[l] * E * NM[l], 256), 256, 0, stream>>>(
        msg[l], dstI, node[l], E, N, NM[l], NP[l]);

  // ---- layers ----
  const int hsh[3] = {5, 4, 3};  // channels-per-head shift for l=0,1,2
  for (int L = 0; L < NLAYERS; ++L) {
    const int pb = LPb(L);
    const float* alpha_W = F(pb + 0);
    const float* alpha_dot = F(pb + 1);
    const float* dstW[3] = {F(pb + 2), F(pb + 3), F(pb + 4)};
    const float* dtp2_w = F(pb + 5);
    const float* ffn_ln_b = F(pb + 6);
    const float* ffn_ln_g[3] = {F(pb + 7), F(pb + 8), F(pb + 9)};
    const float* ffn_post[3] = {F(pb + 10), F(pb + 11), F(pb + 12)};
    const float* ffn_pre[3] = {F(pb + 13), F(pb + 14), F(pb + 15)};
    const float* ln_b = F(pb + 16);
    const float* ln_g[3] = {F(pb + 17), F(pb + 18), F(pb + 19)};
    const float* projW[3] = {F(pb + 20), F(pb + 21), F(pb + 22)};
    const float* rbf_W = F(pb + 23);
    const float* srcW[3] = {F(pb + 24), F(pb + 25), F(pb + 26)};
    const float* val_W[3] = {F(pb + 27), F(pb + 28), F(pb + 29)};
    const float* vlin_W[3] = {F(pb + 30), F(pb + 31), F(pb + 32)};

    // --- attention block ---
    for (int l = 0; l < 3; ++l)
      o3_ln_kernel<<<N, 128, 0, stream>>>(node[l], nn[l], ln_g[l], l == 0 ? ln_b : nullptr,
                                          N, NM[l], NP[l], l == 0);
    for (int l = 0; l < 3; ++l) {
      gemm(nn[l], srcW[l], sb[l], NP[l] * N, NM[l], NM[l], 0);
      gemm(nn[l], dstW[l], db[l], NP[l] * N, NM[l], NM[l], 0);
    }
    for (int l = 0; l < 3; ++l)
      msg_gather_kernel<<<grid1((size_t)NP[l] * E * NM[l], 256), 256, 0, stream>>>(
          sb[l], db[l], srcI, dstI, msg[l], E, N, NM[l], NP[l]);
    gemm(rbfB, rbf_W, w1, E, NBASIS, WNUM, 0);
    dtp(msg[0], msg[1], msg[2], w1, 1, 0, mb[0], mb[1], mb[2]);
    // attention logits + segment softmax over dst
    gemm(mb[0], alpha_W, alphaE, E, 224, 128, 0);
    alpha_prep_kernel<<<grid1((size_t)E * NHEADS, 256), 256, 0, stream>>>(alphaE, alpha_dot, aH, E);
    fill_u_kernel<<<grid1((size_t)N * NHEADS, 256), 256, 0, stream>>>(amax, 0u, (size_t)N * NHEADS);
    amax_kernel<<<grid1((size_t)E * NHEADS, 256), 256, 0, stream>>>(aH, dstI, amax, E);
    fillf(den, 0.f, (size_t)N * NHEADS);
    ea_kernel<<<grid1((size_t)E * NHEADS, 256), 256, 0, stream>>>(aH, dstI, amax, eaB, den, E);
    anorm_kernel<<<grid1((size_t)E * NHEADS, 256), 256, 0, stream>>>(eaB, dstI, den, aH, E);
    // values: gate(o3_linear(m, val_W)) -> msg buffers (msg is dead now)
    for (int l = 0; l < 3; ++l) gemm(mb[l], val_W[l], msg[l], NP[l] * E, DT[l], NM[l], 0);
    sigmoid_kernel<<<grid1((size_t)E * 128, 256), 256, 0, stream>>>(msg[0], (size_t)E * 128);
    gate_vec_kernel<<<grid1((size_t)E * 64, 256), 256, 0, stream>>>(msg[1], E, 64, 3);
    gate_vec_kernel<<<grid1((size_t)E * 32, 256), 256, 0, stream>>>(msg[2], E, 32, 5);
    // dtp2 with shared weights -> mb (mb is dead now), then vlin
    dtp(msg[0], msg[1], msg[2], dtp2_w, /*per-edge*/0, 0, mb[0], mb[1], mb[2]);
    for (int l = 0; l < 3; ++l) gemm(mb[l], vlin_W[l], vout[l], NP[l] * E, DT[l], NM[l], 0);
    // alpha-weighted aggregation into nodes
    for (int l = 0; l < 3; ++l) fillf(att[l], 0.f, (size_t)NP[l] * N * NM[l]);
    for (int l = 0; l < 3; ++l)
      att_scatter_kernel<<<grid1((size_t)NP[l] * E * NM[l], 256), 256, 0, stream>>>(
          vout[l], aH, dstI, att[l], E, N, NM[l], NP[l], hsh[l]);
    // output proj with fused residual add
    for (int l = 0; l < 3; ++l) gemm(att[l], projW[l], node[l], NP[l] * N, NM[l], NM[l], 1);

    // --- FFN block ---
    for (int l = 0; l < 3; ++l)
      o3_ln_kernel<<<N, 128, 0, stream>>>(node[l], nn[l], ffn_ln_g[l], l == 0 ? ffn_ln_b : nullptr,
                                          N, NM[l], NP[l], l == 0);
    for (int l = 0; l < 3; ++l) gemm(nn[l], ffn_pre[l], hid[l], NP[l] * N, NM[l], HM[l], 0);
    sigmoid_kernel<<<grid1((size_t)N * 384, 256), 256, 0, stream>>>(hid[0], (size_t)N * 384);
    gate_vec_kernel<<<grid1((size_t)N * 192, 256), 256, 0, stream>>>(hid[1], N, 192, 3);
    gate_vec_kernel<<<grid1((size_t)N * 96, 256), 256, 0, stream>>>(hid[2], N, 96, 5);
    for (int l = 0; l < 3; ++l) gemm(hid[l], ffn_post[l], node[l], NP[l] * N, HM[l], NM[l], 1);
  }

  // ---- head (only l=0 reaches the scalar output) ----
  for (int l = 0; l < 3; ++l)
    o3_ln_kernel<<<N, 128, 0, stream>>>(node[l], nn[l], h_ln_g[l], l == 0 ? h_ln_b : nullptr,
                                        N, NM[l], NP[l], l == 0);
  gemm(nn[0], h_lin1_0, sb[0], N, 128, 128, 0);
  sigmoid_kernel<<<grid1((size_t)N * 128, 256), 256, 0, stream>>>(sb[0], (size_t)N * 128);
  fillf((float*)d_out, 0.f, (size_t)out_size);
  head_kernel<<<N, 128, 0, stream>>>(sb[0], h_lin2, h_shifts, at_no, batch, (float*)d_out, N);
}